// TransformedGridLoss_20529943674883
// MI455X (gfx1250) — compile-verified
//
#include <hip/hip_runtime.h>

// ---------------------------------------------------------------------------
// TransformedGridLoss, closed form.
//
// Grid moments (axis = linspace(-1,1,64), meshgrid):
//   sum x = sum y = sum xy = 0
//   sum x^2 = sum y^2 = 64 * sum_k (2k/63 - 1)^2 = 5591040/3969
// =>  sum_n ||(A1-A2) p_n + (t1-t2)||^2 = S*||dA||_F^2 + N*||dt||^2
// GT inverse is closed form since the homogeneous bottom row is [0,0,1].
//
// Work: B * ~120 FLOPs, 245 KB read, 4 B written -> launch-latency bound on
// MI455X; one WGP-resident block is optimal. Final reduction uses
// v_wmma_f32_16x16x4_f32 (A = lane partials, B = ones) as an exact wave sum.
// ---------------------------------------------------------------------------

typedef __attribute__((ext_vector_type(2))) float v2f;
typedef __attribute__((ext_vector_type(8))) float v8f;

__device__ __forceinline__ float pair_d2(const float u[6], const float w[6],
                                         float S, float Nf) {
    float d0 = u[0] - w[0], d1 = u[1] - w[1];
    float d3 = u[3] - w[3], d4 = u[4] - w[4];
    float t0 = u[2] - w[2], t1 = u[5] - w[5];
    float dA = d0 * d0 + d1 * d1 + d3 * d3 + d4 * d4;
    float dt = t0 * t0 + t1 * t1;
    return S * dA + Nf * dt;
}

__global__ __launch_bounds__(256) void tg_loss_kernel(
    const float* __restrict__ thAB, const float* __restrict__ thBA,
    const float* __restrict__ thAC, const float* __restrict__ thCA,
    const float* __restrict__ thGT, float* __restrict__ out, int B) {
    const float S  = 5591040.0f / 3969.0f;  // 64 * sum(linspace(-1,1,64)^2)
    const float Nf = 4096.0f;               // grid points per batch

    const int tid = threadIdx.x;
    float acc = 0.0f;

    for (int b = tid; b < B; b += 256) {
        const float* g = thGT + 6 * b;
        float g0 = g[0], g1 = g[1], g2 = g[2];
        float g3 = g[3], g4 = g[4], g5 = g[5];

        // inv([[g0,g1,g2],[g3,g4,g5],[0,0,1]]) -> first two rows
        float rdet = 1.0f / (g0 * g4 - g1 * g3);
        float gi[6];
        gi[0] =  g4 * rdet;
        gi[1] = -g1 * rdet;
        gi[2] = (g1 * g5 - g2 * g4) * rdet;
        gi[3] = -g3 * rdet;
        gi[4] =  g0 * rdet;
        gi[5] = (g2 * g3 - g0 * g5) * rdet;

        float gt[6] = {g0, g1, g2, g3, g4, g5};

        float ab[6], ba[6], ac[6], ca[6];
#pragma unroll
        for (int i = 0; i < 6; ++i) {
            ab[i] = thAB[6 * b + i];
            ba[i] = thBA[6 * b + i];
            ac[i] = thAC[6 * b + i];
            ca[i] = thCA[6 * b + i];
        }

        float l_orig = pair_d2(ab, gt, S, Nf) + pair_d2(ba, gi, S, Nf);
        float l_jit  = pair_d2(ac, gt, S, Nf) + pair_d2(ca, gi, S, Nf);
        float l_id   = pair_d2(ab, ac, S, Nf) + pair_d2(ba, ca, S, Nf);

        acc += 0.5f * l_orig + 0.3f * l_jit + 0.2f * l_id;
    }

    // ---- wave32 sum via V_WMMA_F32_16X16X4_F32 ----------------------------
    // A (16x4 f32): lane L -> A[L%16][2*(L/16)] = acc, other K entry = 0.
    // B = ones  =>  D[m][n] = acc_m + acc_{m+16} (all n identical).
    // C/D layout: lane<16 holds rows v=0..7, lane>=16 rows v+8, so summing the
    // 8 D VGPRs and xor-shuffling across the half-wave yields the exact sum.
    v2f a;
    a.x = acc;
    a.y = 0.0f;
    v2f ones;
    ones.x = 1.0f;
    ones.y = 1.0f;
    v8f c = {};
    c = __builtin_amdgcn_wmma_f32_16x16x4_f32(
        /*neg_a=*/false, a, /*neg_b=*/false, ones,
        /*c_mod=*/(short)0, c, /*reuse_a=*/false, /*reuse_b=*/false);

    float x = c[0] + c[1] + c[2] + c[3] + c[4] + c[5] + c[6] + c[7];
    x += __shfl_xor(x, 16, 32);  // combine the two half-wave partial sums

    __shared__ float wsum[8];
    const int wave = tid >> 5;
    const int lane = tid & 31;
    if (lane == 0) wsum[wave] = x;
    __syncthreads();

    if (tid == 0) {
        float t = 0.0f;
#pragma unroll
        for (int i = 0; i < 8; ++i) t += wsum[i];
        out[0] = t / (static_cast<float>(B) * Nf);
    }
}

extern "C" void kernel_launch(void* const* d_in, const int* in_sizes, int n_in,
                              void* d_out, int out_size, void* d_ws, size_t ws_size,
                              hipStream_t stream) {
    (void)n_in; (void)out_size; (void)d_ws; (void)ws_size;
    const float* thAB = (const float*)d_in[0];
    const float* thBA = (const float*)d_in[1];
    const float* thAC = (const float*)d_in[2];
    const float* thCA = (const float*)d_in[3];
    const float* thGT = (const float*)d_in[4];
    float* out = (float*)d_out;
    const int B = in_sizes[0] / 6;

    // Single WGP-resident block: the problem is launch-latency bound
    // (245 KB in, 4 B out); deterministic single-pass reduction.
    tg_loss_kernel<<<1, 256, 0, stream>>>(thAB, thBA, thAC, thCA, thGT, out, B);
}